// RPN3DPostProcessing_46497315946949
// MI455X (gfx1250) — compile-verified
//
#include <hip/hip_runtime.h>
#include <hip/hip_bf16.h>
#include <stdint.h>

// Problem constants (B=4, C=3, D=H=W=128)
static constexpr unsigned kDHW     = 128u * 128u * 128u;   // 2,097,152
static constexpr unsigned kN       = 4u * kDHW;            // 8,388,608 voxels
static constexpr unsigned kThreads = 256;                  // 8 waves (wave32)
static constexpr unsigned kItems   = 32;                   // elements / thread
static constexpr unsigned kChunk   = kThreads * kItems;    // 8192 elements / block
static constexpr unsigned kNBlocks = kN / kChunk;          // 1024 blocks (exact)
static constexpr float    kThresh  = 0.5f;

// ---------------------------------------------------------------------------
// CDNA5 async VMEM->LDS path (ASYNCcnt-tracked), per CDNA5 ISA
// GV mode: per-lane 64-bit global address in VGPR pair, LDS byte address in VGPR.
// ---------------------------------------------------------------------------
__device__ __forceinline__ void async_load_b128_to_lds(unsigned lds_byte, const void* g) {
    asm volatile("global_load_async_to_lds_b128 %0, %1, off"
                 :: "v"(lds_byte), "v"(g) : "memory");
}
__device__ __forceinline__ void async_wait_all() {
    asm volatile("s_wait_asynccnt 0x0" ::: "memory");
}

// Stage kChunk floats g[start .. start+kChunk) into LDS buffer s_sc (32 KB).
// 8 x b128 per lane: each instruction moves 16B/lane, 512B/wave, contiguous.
__device__ __forceinline__ void stage_scores(float* s_sc, const float* __restrict__ g,
                                             unsigned start, unsigned tid) {
#pragma unroll
    for (unsigned j = 0; j < kChunk / (kThreads * 4u); ++j) {
        unsigned e   = (j * kThreads + tid) * 4u;                  // float idx in chunk
        unsigned lds = (unsigned)(uintptr_t)(&s_sc[e]);            // low 32b = LDS byte addr
        async_load_b128_to_lds(lds, (const void*)(g + start + e)); // 16B-aligned
    }
    async_wait_all();
    __syncthreads();
}

// ---------------------------------------------------------------------------
// Pass 1: per-block count of passing scores
// ---------------------------------------------------------------------------
__global__ void rpn_count_kernel(const float* __restrict__ scores,
                                 unsigned* __restrict__ counts) {
    __shared__ __align__(16) float s_sc[kChunk];
    __shared__ unsigned s_cnt;
    const unsigned tid   = threadIdx.x;
    const unsigned start = blockIdx.x * kChunk;
    if (tid == 0) s_cnt = 0u;                // visible after barrier inside stage_scores
    stage_scores(s_sc, scores, start, tid);

    unsigned c = 0;
#pragma unroll
    for (unsigned j = 0; j < kItems; ++j)
        c += (s_sc[j * kThreads + tid] > kThresh) ? 1u : 0u;

    atomicAdd(&s_cnt, c);
    __syncthreads();
    if (tid == 0) counts[blockIdx.x] = s_cnt;
}

// ---------------------------------------------------------------------------
// Pass 2: exclusive scan of the 1024 block counts (single 1024-thread block)
// ws[0..1023]=counts -> ws[1024..2047]=exclusive offsets, ws[2048]=total
// ---------------------------------------------------------------------------
__global__ void rpn_scan_kernel(unsigned* __restrict__ ws) {
    __shared__ unsigned sd[kNBlocks];
    const unsigned tid = threadIdx.x;            // blockDim.x == kNBlocks == 1024
    const unsigned v   = ws[tid];
    sd[tid] = v;
    __syncthreads();
    for (unsigned off = 1; off < kNBlocks; off <<= 1) {
        unsigned t = (tid >= off) ? sd[tid - off] : 0u;
        __syncthreads();
        sd[tid] += t;
        __syncthreads();
    }
    const unsigned incl = sd[tid];
    ws[kNBlocks + tid] = incl - v;               // exclusive prefix
    if (tid == kNBlocks - 1) ws[2 * kNBlocks] = incl;  // grand total
}

// ---------------------------------------------------------------------------
// Pass 3: order-preserving scatter of passing coords into points[0:count]
// ---------------------------------------------------------------------------
__global__ void rpn_scatter_kernel(const float* __restrict__ coords,
                                   const float* __restrict__ scores,
                                   const unsigned* __restrict__ ws,
                                   float* __restrict__ out) {
    __shared__ __align__(16) float s_sc[kChunk];
    __shared__ unsigned s_w[kThreads / 32u];     // per-wave pass counts (8 waves)
    const unsigned tid   = threadIdx.x;
    const unsigned lane  = tid & 31u;
    const unsigned wid   = tid >> 5;
    const unsigned start = blockIdx.x * kChunk;
    unsigned running = ws[kNBlocks + blockIdx.x];

    // Prefetch the 3 coord planes for this block's range (global_prefetch_b8)
    {
        const unsigned idx = start + tid;
        const unsigned b = idx >> 21, s = idx & (kDHW - 1u);
        __builtin_prefetch(coords + (((size_t)(b * 3u + 0u)) << 21) + s, 0, 1);
        __builtin_prefetch(coords + (((size_t)(b * 3u + 1u)) << 21) + s, 0, 1);
        __builtin_prefetch(coords + (((size_t)(b * 3u + 2u)) << 21) + s, 0, 1);
    }

    stage_scores(s_sc, scores, start, tid);

    for (unsigned j = 0; j < kItems; ++j) {
        const unsigned k   = j * kThreads + tid;
        const unsigned idx = start + k;
        const bool pred = s_sc[k] > kThresh;

        const unsigned long long m = __ballot(pred);     // wave32: low 32 bits
        const unsigned wcnt = (unsigned)__popcll(m);
        if (lane == 0) s_w[wid] = wcnt;
        __syncthreads();

        unsigned wpre = 0, tot = 0;
#pragma unroll
        for (unsigned w = 0; w < kThreads / 32u; ++w) {
            const unsigned c = s_w[w];
            wpre += (w < wid) ? c : 0u;
            tot  += c;
        }
        const unsigned lpre = (unsigned)__popcll(m & ((1ull << lane) - 1ull));

        if (pred) {
            const unsigned pos = running + wpre + lpre;
            const unsigned b = idx >> 21, s = idx & (kDHW - 1u);
            const size_t base = (((size_t)(b * 3u)) << 21) + s;
            const float x = coords[base];
            const float y = coords[base + (size_t)kDHW];
            const float z = coords[base + 2u * (size_t)kDHW];
            const size_t o = (size_t)pos * 3u;
            out[o + 0] = x; out[o + 1] = y; out[o + 2] = z;
        }
        running += tot;
        __syncthreads();                       // protect s_w for next iteration
    }
}

// ---------------------------------------------------------------------------
// Pass 4: valid[] step function, zero the points tail, write count
// ---------------------------------------------------------------------------
__global__ void rpn_tail_kernel(const unsigned* __restrict__ ws,
                                float* __restrict__ out) {
    const unsigned total = ws[2 * kNBlocks];
    const unsigned j0 = (blockIdx.x * kThreads + threadIdx.x) * 4u;

    float4 v;
    v.x = (j0 + 0u < total) ? 1.f : 0.f;
    v.y = (j0 + 1u < total) ? 1.f : 0.f;
    v.z = (j0 + 2u < total) ? 1.f : 0.f;
    v.w = (j0 + 3u < total) ? 1.f : 0.f;
    *reinterpret_cast<float4*>(out + (size_t)3u * kN + j0) = v;

#pragma unroll
    for (unsigned t = 0; t < 4u; ++t) {
        const unsigned j = j0 + t;
        if (j >= total) {
            const size_t o = (size_t)j * 3u;
            out[o + 0] = 0.f; out[o + 1] = 0.f; out[o + 2] = 0.f;
        }
    }
    if (j0 == 0u) out[(size_t)4u * kN] = (float)total;
}

// ---------------------------------------------------------------------------
extern "C" void kernel_launch(void* const* d_in, const int* in_sizes, int n_in,
                              void* d_out, int out_size, void* d_ws, size_t ws_size,
                              hipStream_t stream) {
    (void)in_sizes; (void)n_in; (void)out_size; (void)ws_size;
    const float* coords = (const float*)d_in[0];   // [4,3,128,128,128] f32
    const float* scores = (const float*)d_in[1];   // [4,1,128,128,128] f32
    float*    out = (float*)d_out;                 // [3N pts][N valid][1 count]
    unsigned* ws  = (unsigned*)d_ws;               // counts | offsets | total

    rpn_count_kernel  <<<kNBlocks, kThreads, 0, stream>>>(scores, ws);
    rpn_scan_kernel   <<<1,        kNBlocks, 0, stream>>>(ws);
    rpn_scatter_kernel<<<kNBlocks, kThreads, 0, stream>>>(coords, scores, ws, out);
    rpn_tail_kernel   <<<kN / (kThreads * 4u), kThreads, 0, stream>>>(ws, out);
}